// ChamferDistanceLoss_48172353192475
// MI455X (gfx1250) — compile-verified
//
#include <hip/hip_runtime.h>

typedef float v2f __attribute__((ext_vector_type(2)));
typedef float v8f __attribute__((ext_vector_type(8)));

#define TILE 16

// Workspace layout (floats):
//   Aaug[nP][4] = { px, py, pz, 1 }
//   Baug[nT][4] = { -2tx, -2ty, -2tz, |t|^2 }
//   p2  [nP]    = |p|^2
//   part[nP/16] = per-pred-tile partial sums of sqrt(min d^2)

__global__ __launch_bounds__(256)
void chamfer_prep(const float* __restrict__ pred,
                  const float* __restrict__ target,
                  float* __restrict__ Aaug,
                  float* __restrict__ Baug,
                  float* __restrict__ p2,
                  int nP, int nT) {
    int i = blockIdx.x * blockDim.x + threadIdx.x;
    if (i < nP) {
        float px = pred[3 * i + 0];
        float py = pred[3 * i + 1];
        float pz = pred[3 * i + 2];
        ((float4*)Aaug)[i] = make_float4(px, py, pz, 1.0f);
        p2[i] = px * px + py * py + pz * pz;
    }
    if (i < nT) {
        float tx = target[3 * i + 0];
        float ty = target[3 * i + 1];
        float tz = target[3 * i + 2];
        ((float4*)Baug)[i] =
            make_float4(-2.0f * tx, -2.0f * ty, -2.0f * tz,
                        tx * tx + ty * ty + tz * tz);
    }
}

// One wave (32 threads) per 16-row pred tile. For each 16-col target tile:
//   D = A(16x4) * B(4x16) + C   where D[i][j] = d^2(i,j)
// A fragment: lane l<16 holds row l, K={0,1}; lanes 16-31 hold row l-16, K={2,3}
// B fragment: lane n<16 holds col n, K={0,1}; lanes 16-31 hold col n-16, K={2,3}
// Both are a single 8B load from the row-major [N][4] augmented arrays.
// C fragment: VGPR v = p2[i0 + v + 8*(lane>=16)]  (constant over the j loop).
__global__ __launch_bounds__(32)
void chamfer_min_wmma(const float* __restrict__ Aaug,
                      const float* __restrict__ Baug,
                      const float* __restrict__ p2,
                      float* __restrict__ partial,
                      int nT) {
    const int lane = threadIdx.x;
    const int half = lane >> 4;      // 0: K pair {0,1}, 1: K pair {2,3}
    const int l    = lane & 15;
    const int i0   = blockIdx.x * TILE;

    // A fragment (fixed for the whole wave's j loop)
    v2f a = *(const v2f*)(Aaug + (size_t)(i0 + l) * 4 + half * 2);

    // C fragment = |p|^2 per row, broadcast across lanes (row-only dependence)
    v8f c;
#pragma unroll
    for (int v = 0; v < 8; ++v) c[v] = p2[i0 + half * 8 + v];

    v8f m0, m1;
#pragma unroll
    for (int v = 0; v < 8; ++v) { m0[v] = 3.0e38f; m1[v] = 3.0e38f; }

    // Stream all target tiles, 2 per iteration for ILP across WMMA issues.
    for (int j0 = 0; j0 < nT; j0 += 2 * TILE) {
        v2f b0 = *(const v2f*)(Baug + (size_t)(j0 + l) * 4 + half * 2);
        v2f b1 = *(const v2f*)(Baug + (size_t)(j0 + TILE + l) * 4 + half * 2);

        v8f d0 = __builtin_amdgcn_wmma_f32_16x16x4_f32(
            false, a, false, b0, (short)0, c, false, false);
        v8f d1 = __builtin_amdgcn_wmma_f32_16x16x4_f32(
            false, a, false, b1, (short)0, c, false, false);

#pragma unroll
        for (int v = 0; v < 8; ++v) {
            m0[v] = fminf(m0[v], d0[v]);
            m1[v] = fminf(m1[v], d1[v]);
        }
    }
#pragma unroll
    for (int v = 0; v < 8; ++v) m0[v] = fminf(m0[v], m1[v]);

    // Butterfly min across the 16 lanes of each half (columns within tile).
#pragma unroll
    for (int mask = 1; mask <= 8; mask <<= 1) {
#pragma unroll
        for (int v = 0; v < 8; ++v) {
            float o = __shfl_xor(m0[v], mask, 32);
            m0[v] = fminf(m0[v], o);
        }
    }

    // Every lane in each half now holds the per-row min d^2 for its 8 rows.
    float s = 0.0f;
#pragma unroll
    for (int v = 0; v < 8; ++v) s += sqrtf(fmaxf(m0[v], 0.0f));

    // lane 0 carries rows 0..7, lane 16 carries rows 8..15
    float s_hi = __shfl(s, 16, 32);
    if (lane == 0) partial[blockIdx.x] = s + s_hi;
}

__global__ __launch_bounds__(256)
void chamfer_finish(const float* __restrict__ partial, float* __restrict__ out,
                    int nTiles, float inv_n) {
    __shared__ float sm[256];
    int t = threadIdx.x;
    float s = 0.0f;
    for (int i = t; i < nTiles; i += 256) s += partial[i];
    sm[t] = s;
    __syncthreads();
    for (int off = 128; off > 0; off >>= 1) {
        if (t < off) sm[t] += sm[t + off];
        __syncthreads();
    }
    if (t == 0) out[0] = sm[0] * inv_n;
}

extern "C" void kernel_launch(void* const* d_in, const int* in_sizes, int n_in,
                              void* d_out, int out_size, void* d_ws, size_t ws_size,
                              hipStream_t stream) {
    const float* pred   = (const float*)d_in[0];
    const float* target = (const float*)d_in[1];
    float* out = (float*)d_out;

    const int nP = in_sizes[0] / 3;   // 16384
    const int nT = in_sizes[1] / 3;   // 16384
    const int nTilesP = nP / TILE;    // 1024

    char* ws = (char*)d_ws;
    float* Aaug = (float*)ws;
    float* Baug = (float*)(ws + (size_t)nP * 16);
    float* p2   = (float*)(ws + (size_t)nP * 16 + (size_t)nT * 16);
    float* part = p2 + nP;

    int nMax = nP > nT ? nP : nT;
    chamfer_prep<<<(nMax + 255) / 256, 256, 0, stream>>>(
        pred, target, Aaug, Baug, p2, nP, nT);

    chamfer_min_wmma<<<nTilesP, 32, 0, stream>>>(Aaug, Baug, p2, part, nT);

    chamfer_finish<<<1, 256, 0, stream>>>(part, out, nTilesP, 1.0f / (float)nP);
}